// FQCE_8933531975808
// MI455X (gfx1250) — compile-verified
//
#include <hip/hip_runtime.h>

// Fully-quantum KGE scoring: one wave32 per (s,p,o) triple.
// 2^5 = 32 amplitudes -> one complex amplitude per lane, state lives in VGPRs.
// Gates = shfl_xor butterfly + 2x2 complex mat-vec with readlane-broadcast coeffs.
// Final Re<sp|o> reduction via V_WMMA_F32_16X16X4_F32 (row-sum trick, layout-robust).

#define NGATES 20

typedef __attribute__((ext_vector_type(2))) float v2f;
typedef __attribute__((ext_vector_type(8))) float v8f;

__device__ __forceinline__ float bcast_lane(float v, int l) {
  return __builtin_bit_cast(float,
      __builtin_amdgcn_readlane(__builtin_bit_cast(int, v), l));
}

// Apply one 20-gate circuit block. gp -> 60 floats = [4][5][3] angles.
// Lane g (g<20) builds gate g's Rot matrix; coefficients broadcast per gate.
__device__ __forceinline__ void apply_block(const float* __restrict__ gp, int lane,
                                            float& ar, float& ai) {
  float phi = 0.f, th = 0.f, om = 0.f;
  if (lane < NGATES) {
    phi = gp[lane * 3 + 0];
    th  = gp[lane * 3 + 1];
    om  = gp[lane * 3 + 2];
  }
  // Rot(phi,theta,omega) = RZ(om) RY(th) RZ(phi)
  // u00 = e^{-i a} c ; u01 = (-C, D) ; u10 = (C, D) ; u11 = (A, -B)
  // with a=(phi+om)/2, b=(phi-om)/2, c=cos(th/2), s=sin(th/2)
  float s, c;    __sincosf(0.5f * th,         &s,  &c);
  float sa, ca;  __sincosf(0.5f * (phi + om), &sa, &ca);
  float sb, cbv; __sincosf(0.5f * (phi - om), &sb, &cbv);
  const float Am =  c * ca;   // Re u00
  const float Bm = -c * sa;   // Im u00
  const float Cm =  s * cbv;  // Re u10
  const float Dm = -s * sb;   // Im u10

  // wire q <-> bit (4-q) of the amplitude index (wire 0 most significant).
  // gates 0..4: 1q Rot on wire q.  gates 5..19: CRot ctrl q, tgt (q+off)%5, off=1,2,3.
  constexpr int TB[NGATES] = {4,3,2,1,0,  3,2,1,0,4,  2,1,0,4,3,  1,0,4,3,2};
  constexpr int CB[NGATES] = {-1,-1,-1,-1,-1, 4,3,2,1,0, 4,3,2,1,0, 4,3,2,1,0};

#pragma unroll
  for (int g = 0; g < NGATES; ++g) {
    const float Ab = bcast_lane(Am, g);
    const float Bb = bcast_lane(Bm, g);
    const float Cb = bcast_lane(Cm, g);
    const float Db = bcast_lane(Dm, g);
    const int tbit = TB[g];
    const float apr = __shfl_xor(ar, 1 << tbit, 32);
    const float api = __shfl_xor(ai, 1 << tbit, 32);
    const bool hi = (lane >> tbit) & 1;           // this lane's target bit
    const float sB = hi ? -Bb : Bb;               // Im of coeff-of-a   (u00 / u11)
    const float tC = hi ?  Cb : -Cb;              // Re of coeff-of-ap  (u01 / u10)
    const float nr = Ab * ar - sB * ai + tC * apr - Db * api;
    const float ni = Ab * ai + sB * ar + tC * api + Db * apr;
    const bool act = (CB[g] < 0) || ((lane >> CB[g]) & 1);  // control bit
    ar = act ? nr : ar;
    ai = act ? ni : ai;
  }
}

__global__ __launch_bounds__(256) void fqce_score_kernel(
    const int* __restrict__ s, const int* __restrict__ p, const int* __restrict__ o,
    const float* __restrict__ ent, const float* __restrict__ rel,
    float* __restrict__ out, int Bn) {
  const int tid  = blockIdx.x * blockDim.x + threadIdx.x;
  const int w    = tid >> 5;   // triple index: one wave per triple
  const int lane = tid & 31;   // amplitude index (wave32: 32 lanes == 2^5 amps)
  if (w >= Bn) return;         // whole-wave uniform -> EXEC stays all-ones

  const int si = __builtin_amdgcn_readfirstlane(s[w]);
  const int pi = __builtin_amdgcn_readfirstlane(p[w]);
  const int oi = __builtin_amdgcn_readfirstlane(o[w]);

  const float inv = 0.17677669529663688f;  // 1/sqrt(32): uniform H^5 |0..0>

  float spr = inv, spi = 0.f;
  apply_block(ent + (long)si * 60, lane, spr, spi);
  apply_block(rel + (long)pi * 60, lane, spr, spi);

  float orr = inv, ori = 0.f;
  apply_block(ent + (long)oi * 60, lane, orr, ori);

  // Re(conj(sp).o) per lane
  float cL = spr * orr + spi * ori;

  float t;
#if defined(__has_builtin) && __has_builtin(__builtin_amdgcn_wmma_f32_16x16x4_f32)
  // Wave reduction via WMMA: A holds cL in its first VGPR slot (second = 0),
  // B = ones => every D row = row-sum of A. Summing a lane's 8 acc VGPRs gives
  // sum over half the rows; shfl_xor(16) completes Sum(cL) in every lane.
  // Correct for ANY bijective (lane,vgpr)<->(m,k) A layout.
  v2f a; a[0] = cL;  a[1] = 0.f;
  v2f b; b[0] = 1.f; b[1] = 1.f;
  v8f acc = {};
  v8f d = __builtin_amdgcn_wmma_f32_16x16x4_f32(
      /*neg_a=*/false, a, /*neg_b=*/false, b,
      /*c_mod=*/(short)0, acc, /*reuse_a=*/false, /*reuse_b=*/false);
  t = ((d[0] + d[1]) + (d[2] + d[3])) + ((d[4] + d[5]) + (d[6] + d[7]));
  t += __shfl_xor(t, 16, 32);
#else
  t = cL;
#pragma unroll
  for (int m = 16; m >= 1; m >>= 1) t += __shfl_xor(t, m, 32);
#endif

  if (lane == 0) out[w] = t;
}

extern "C" void kernel_launch(void* const* d_in, const int* in_sizes, int n_in,
                              void* d_out, int out_size, void* d_ws, size_t ws_size,
                              hipStream_t stream) {
  (void)n_in; (void)d_ws; (void)ws_size; (void)out_size;
  const int*   s   = (const int*)d_in[0];
  const int*   p   = (const int*)d_in[1];
  const int*   o   = (const int*)d_in[2];
  const float* ent = (const float*)d_in[3];
  const float* rel = (const float*)d_in[4];
  float* out = (float*)d_out;

  const int Bn = in_sizes[0];                 // 65536 triples
  const long total = (long)Bn * 32;           // one wave32 per triple
  const int threads = 256;                    // 8 waves per block
  const int blocks = (int)((total + threads - 1) / threads);

  fqce_score_kernel<<<blocks, threads, 0, stream>>>(s, p, o, ent, rel, out, Bn);
}